// Adaptive_Perspective_Transformation_44590350467284
// MI455X (gfx1250) — compile-verified
//
#include <hip/hip_runtime.h>

typedef float v2f __attribute__((ext_vector_type(2)));
typedef float v8f __attribute__((ext_vector_type(8)));

#ifndef __has_builtin
#define __has_builtin(x) 0
#endif

constexpr int kN = 8, kC = 16, kH = 512, kW = 512;
constexpr int kTilesX = kW / 16, kTilesY = kH / 16;
constexpr int kWavesPerBlock = 8;

// ---------------------------------------------------------------------------
// Kernel 1: per-batch homography solve (8x8 Gaussian elimination w/ partial
// pivoting), 3x3 inversion, and folding of the grid-normalization constants:
//   ix = (Minv@[x,y,1])_x / w * W/(W-1) - 0.5
// is folded into row0' = s_x*row0 - 0.5*row2 (same for row1), so the warp
// kernel computes ix = u'/w directly. 8 threads total -> negligible cost.
// ---------------------------------------------------------------------------
__global__ void solve_minv_kernel(const float* __restrict__ co,
                                  float* __restrict__ ws) {
  int n = threadIdx.x;
  if (n >= kN) return;
  const float iw = (float)kW, ih = (float)kH;
  // Faithful to reference: corner = [[0,0],[0,ih],[iw,0],[ih,iw]] as (x,y)
  float sx[4] = {0.f, 0.f, iw, ih};
  float sy[4] = {0.f, ih, 0.f, iw};

  float A[8][9];
  for (int p = 0; p < 4; ++p) {
    float dx = sx[p] + co[n * 8 + 2 * p + 0];
    float dy = sy[p] + co[n * 8 + 2 * p + 1];
    float* e = A[2 * p];
    float* o = A[2 * p + 1];
    e[0] = sx[p]; e[1] = sy[p]; e[2] = 1.f; e[3] = 0.f; e[4] = 0.f; e[5] = 0.f;
    e[6] = -dx * sx[p]; e[7] = -dx * sy[p]; e[8] = dx;
    o[0] = 0.f; o[1] = 0.f; o[2] = 0.f; o[3] = sx[p]; o[4] = sy[p]; o[5] = 1.f;
    o[6] = -dy * sx[p]; o[7] = -dy * sy[p]; o[8] = dy;
  }
  // Gaussian elimination with partial pivoting
  for (int k = 0; k < 8; ++k) {
    int piv = k;
    float best = fabsf(A[k][k]);
    for (int i = k + 1; i < 8; ++i) {
      float v = fabsf(A[i][k]);
      if (v > best) { best = v; piv = i; }
    }
    if (piv != k) {
      for (int j = k; j < 9; ++j) {
        float t = A[k][j]; A[k][j] = A[piv][j]; A[piv][j] = t;
      }
    }
    float inv = 1.f / A[k][k];
    for (int i = k + 1; i < 8; ++i) {
      float f = A[i][k] * inv;
      for (int j = k; j < 9; ++j) A[i][j] -= f * A[k][j];
    }
  }
  float h[8];
  for (int k = 7; k >= 0; --k) {
    float s = A[k][8];
    for (int j = k + 1; j < 8; ++j) s -= A[k][j] * h[j];
    h[k] = s / A[k][k];
  }
  // M and its inverse (adjugate / det)
  float m00 = h[0], m01 = h[1], m02 = h[2];
  float m10 = h[3], m11 = h[4], m12 = h[5];
  float m20 = h[6], m21 = h[7], m22 = 1.f;
  float c00 = m11 * m22 - m12 * m21;
  float c10 = -(m10 * m22 - m12 * m20);
  float c20 = m10 * m21 - m11 * m20;
  float det = m00 * c00 + m01 * c10 + m02 * c20;
  float id = 1.f / det;
  float i00 = c00 * id;
  float i01 = (m02 * m21 - m01 * m22) * id;
  float i02 = (m01 * m12 - m02 * m11) * id;
  float i10 = c10 * id;
  float i11 = (m00 * m22 - m02 * m20) * id;
  float i12 = (m02 * m10 - m00 * m12) * id;
  float i20 = c20 * id;
  float i21 = (m01 * m20 - m00 * m21) * id;
  float i22 = (m00 * m11 - m01 * m10) * id;

  float sxs = iw / (iw - 1.f);
  float sys = ih / (ih - 1.f);
  float* m = ws + n * 12;
  m[0] = sxs * i00 - 0.5f * i20;
  m[1] = sxs * i01 - 0.5f * i21;
  m[2] = sxs * i02 - 0.5f * i22;
  m[3] = sys * i10 - 0.5f * i20;
  m[4] = sys * i11 - 0.5f * i21;
  m[5] = sys * i12 - 0.5f * i22;
  m[6] = i20; m[7] = i21; m[8] = i22;
}

// ---------------------------------------------------------------------------
// Kernel 2: one wave32 per 16x16 output tile. Three V_WMMA_F32_16X16X4_F32
// produce the 16x16 fields of homogeneous numerators U,V and denominator W:
//   A[r] = [y_r, 1, 0, 0]   (16x4; lanes 16-31 hold K=2,3 -> zero)
//   B[:,c] = [coef_y, coef_x*x_c + coef_1, 0, 0]
//   D[r][c] = coef_y*y_r + coef_x*x_c + coef_1
// C/D layout: VGPR j = row j (lanes 0-15) / row j+8 (lanes 16-31), col = lane&15.
// Each lane then bilinear-samples its 8 pixels x 16 channels; taps hit L2
// (input 134MB < 192MB L2); output goes out via non-temporal stores.
// ---------------------------------------------------------------------------
__global__ __launch_bounds__(256) void warp_kernel(
    const float* __restrict__ img, const float* __restrict__ ws,
    float* __restrict__ out) {
  const int lane = threadIdx.x & 31;
  const int wave = threadIdx.x >> 5;
  const int tile = blockIdx.x * kWavesPerBlock + wave;
  if (tile >= kN * kTilesX * kTilesY) return;  // wave-uniform; EXEC stays full
  const int n = tile / (kTilesX * kTilesY);
  const int tr = tile % (kTilesX * kTilesY);
  const int ty = tr / kTilesX;
  const int tx = tr % kTilesX;

  const float x0f = (float)(tx * 16);
  const float y0f = (float)(ty * 16);
  const float* __restrict__ m = ws + n * 12;
  const bool hi = lane >= 16;
  const int col = lane & 15;
  const float xc = x0f + (float)col;

  // A fragment (shared by all three WMMAs)
  v2f a;
  a.x = hi ? 0.f : (y0f + (float)lane);  // K=0 (lo lanes) / K=2 (hi lanes)=0
  a.y = hi ? 0.f : 1.f;                  // K=1 / K=3=0
  // B fragments: row K=0 = y-coefficient, row K=1 = coef_x*x_c + coef_1
  v2f bu, bv, bw;
  bu.x = hi ? 0.f : m[1];
  bu.y = hi ? 0.f : fmaf(m[0], xc, m[2]);
  bv.x = hi ? 0.f : m[4];
  bv.y = hi ? 0.f : fmaf(m[3], xc, m[5]);
  bw.x = hi ? 0.f : m[7];
  bw.y = hi ? 0.f : fmaf(m[6], xc, m[8]);

  v8f z = {};
  v8f U, V, Wd;
#if __has_builtin(__builtin_amdgcn_wmma_f32_16x16x4_f32)
  U = __builtin_amdgcn_wmma_f32_16x16x4_f32(false, a, false, bu, (short)0, z,
                                            false, false);
  V = __builtin_amdgcn_wmma_f32_16x16x4_f32(false, a, false, bv, (short)0, z,
                                            false, false);
  Wd = __builtin_amdgcn_wmma_f32_16x16x4_f32(false, a, false, bw, (short)0, z,
                                             false, false);
#else
  #pragma unroll
  for (int j = 0; j < 8; ++j) {
    float yr = y0f + (float)(j + (hi ? 8 : 0));
    U[j]  = fmaf(m[1], yr, fmaf(m[0], xc, m[2]));
    V[j]  = fmaf(m[4], yr, fmaf(m[3], xc, m[5]));
    Wd[j] = fmaf(m[7], yr, fmaf(m[6], xc, m[8]));
  }
#endif

  const int xdst = tx * 16 + col;
  const int imgN = n * kC * kH * kW;
  #pragma unroll
  for (int j = 0; j < 8; ++j) {
    const int ydst = ty * 16 + j + (hi ? 8 : 0);
    const float rw = __builtin_amdgcn_rcpf(Wd[j]);
    const float ix = U[j] * rw;  // scale & -0.5 already folded in
    const float iy = V[j] * rw;
    const float fx = floorf(ix);
    const float fy = floorf(iy);
    const float wx = ix - fx;
    const float wy = iy - fy;
    const int ix0 = (int)fx, iy0 = (int)fy;
    const int ix1 = ix0 + 1, iy1 = iy0 + 1;
    const bool vx0 = (ix0 >= 0) & (ix0 < kW);
    const bool vx1 = (ix1 >= 0) & (ix1 < kW);
    const bool vy0 = (iy0 >= 0) & (iy0 < kH);
    const bool vy1 = (iy1 >= 0) & (iy1 < kH);
    const int cx0 = min(max(ix0, 0), kW - 1);
    const int cx1 = min(max(ix1, 0), kW - 1);
    const int cy0 = min(max(iy0, 0), kH - 1);
    const int cy1 = min(max(iy1, 0), kH - 1);
    const float w00 = (1.f - wx) * (1.f - wy) * ((vx0 & vy0) ? 1.f : 0.f);
    const float w01 = wx * (1.f - wy) * ((vx1 & vy0) ? 1.f : 0.f);
    const float w10 = (1.f - wx) * wy * ((vx0 & vy1) ? 1.f : 0.f);
    const float w11 = wx * wy * ((vx1 & vy1) ? 1.f : 0.f);
    const int r0 = cy0 * kW;
    const int r1 = cy1 * kW;
    const int odst = ydst * kW + xdst;
    #pragma unroll
    for (int c = 0; c < kC; ++c) {
      const int cb = imgN + c * (kH * kW);
      const float t00 = img[cb + r0 + cx0];
      const float t01 = img[cb + r0 + cx1];
      const float t10 = img[cb + r1 + cx0];
      const float t11 = img[cb + r1 + cx1];
      float acc = t00 * w00;
      acc = fmaf(t01, w01, acc);
      acc = fmaf(t10, w10, acc);
      acc = fmaf(t11, w11, acc);
      __builtin_nontemporal_store(acc, &out[cb + odst]);
    }
  }
}

extern "C" void kernel_launch(void* const* d_in, const int* in_sizes, int n_in,
                              void* d_out, int out_size, void* d_ws,
                              size_t ws_size, hipStream_t stream) {
  const float* x = (const float*)d_in[0];       // (8,16,512,512) f32
  const float* co = (const float*)d_in[1];      // (8,8) f32
  float* out = (float*)d_out;                   // (8,16,512,512) f32
  float* ws = (float*)d_ws;                     // 8*12 floats of coeffs

  solve_minv_kernel<<<1, kN, 0, stream>>>(co, ws);

  const int totalTiles = kN * kTilesX * kTilesY;          // 8192 waves
  const int blocks = totalTiles / kWavesPerBlock;         // 1024 blocks x 256T
  warp_kernel<<<blocks, 256, 0, stream>>>(x, ws, out);
}